// Bihomogeneous_k3_25537875542467
// MI455X (gfx1250) — compile-verified
//
#include <hip/hip_runtime.h>
#include <cstdint>
#include <cstddef>

// ---------------------------------------------------------------------------
// Bihomogeneous k=3 moment kernel for MI455X (gfx1250, wave32).
//
// Per row b (B=131072): zz[35] = z_i*z_j*z_k (i<=j<=k over 5 complex values),
// out[b, 0..629]    = Re(zz_p * conj(zz_q))  for pairs p<=q (p-major order)
// out[b, 630..1224] = Im(zz_p * conj(zz_q))  for strict pairs p<q, same order.
//
// Roofline: 642 MB f32 output vs 5.2 MB input -> store-bound, ~27.5 us floor
// at 23.3 TB/s. No contraction dimension exists (per-row rank-2 Gram), so
// WMMA is intentionally not used; the goal is minimum issue slots per stored
// byte:
//   - one wave32 per TWO adjacent rows (16 rows / 256-thread block): the
//     column table load, offset unpack and LDS address adds are shared; the
//     second row rides on immediate offsets (+280 B in LDS, +4900 B in the
//     output), costing only its ds_load_b64s, FMAs and store.
//   - CDNA5 async global->LDS staging of inputs (global_load_async_to_lds_b32)
//   - zz as interleaved float2 in LDS -> one ds_load_b64 per operand
//   - output columns flattened into 39 DENSE wave-iterations; only the last
//     (9 lanes) is predicated, only iteration 19 mixes re/im.
//   - non-temporal stores: 642 MB write-once must not thrash the 192 MB L2.
// ---------------------------------------------------------------------------

#define ROWS_PER_BLOCK 16   // 8 waves x 2 rows
#define NTRI 35             // monomials (i<=j<=k, n=5, k=3)
#define NRE  630            // pairs p<=q
#define NCOL 1225           // 630 re + 595 strict im
#define NITER 39            // ceil(1225 / 32)

struct TriTab { unsigned v[NTRI]; };
constexpr TriTab make_tri() {
  TriTab t{}; int m = 0;
  for (int i = 0; i < 5; ++i)
    for (int j = i; j < 5; ++j)
      for (int k = j; k < 5; ++k)
        t.v[m++] = (unsigned)(i | (j << 4) | (k << 8));
  return t;
}
__constant__ TriTab TRI = make_tri();

// Per-output-column table: low16 = p*8 (LDS byte offset of zz[p]),
// high16 = q*8. Columns 0..629: re pairs (p-major, q>=p); 630..1224:
// strict-im pairs (p-major, q>p). Padded to 39*32 for the masked tail.
struct ColTab { unsigned v[NITER * 32]; };
constexpr ColTab make_cols() {
  ColTab t{}; int c = 0;
  for (int p = 0; p < NTRI; ++p)
    for (int q = p; q < NTRI; ++q)
      t.v[c++] = (unsigned)((p * 8) | ((q * 8) << 16));
  for (int p = 0; p < NTRI; ++p)
    for (int q = p + 1; q < NTRI; ++q)
      t.v[c++] = (unsigned)((p * 8) | ((q * 8) << 16));
  while (c < NITER * 32) t.v[c++] = 0;
  return t;
}
__constant__ ColTab COLTAB = make_cols();

typedef __attribute__((address_space(3))) float lds_float;

// Hermitian pair value for one row given the (a,b) operands of this column.
__device__ __forceinline__ float pair_val(int it, int lane, float2 a, float2 b) {
  if (it < 19) {                         // pure re region (c <= 607)
    return fmaf(a.x, b.x, a.y * b.y);
  } else if (it == 19) {                 // mixed: c = 608+lane; im iff lane>=22
    const bool isIm = lane >= 22;
    const float s1 = isIm ? a.y : a.x;
    const float s2 = isIm ? -a.x : a.y;
    return fmaf(s1, b.x, s2 * b.y);
  } else {                               // pure im region
    return fmaf(a.y, b.x, -(a.x * b.y));
  }
}

template <bool TWO>
__device__ __forceinline__ void sweep(const char* zzb, float* outRow, int lane) {
#pragma unroll
  for (int it = 0; it < NITER; ++it) {
    const unsigned tbl = COLTAB.v[it * 32 + lane]; // saddr + imm offset load
    const unsigned poff = tbl & 0xFFFFu;
    const unsigned qoff = tbl >> 16;
    const char* pa = zzb + poff;                   // one v_add each; row 1
    const char* pb = zzb + qoff;                   // rides on imm offsets
    const float2 a0 = *(const float2*)(pa);
    const float2 b0 = *(const float2*)(pb);
    float2 a1, b1;
    if (TWO) {
      a1 = *(const float2*)(pa + NTRI * 8);        // ds imm offset +280
      b1 = *(const float2*)(pb + NTRI * 8);
    }
    const float v0 = pair_val(it, lane, a0, b0);
    float v1 = 0.0f;
    if (TWO) v1 = pair_val(it, lane, a1, b1);
    const int off = it * 32 + lane;                // folds to store imm
    if (it < NITER - 1 || lane < NCOL - (NITER - 1) * 32) {
      __builtin_nontemporal_store(v0, outRow + off);
      if (TWO) __builtin_nontemporal_store(v1, outRow + NCOL + off);
    }
  }
}

__global__ __launch_bounds__(256) void bihom_k3_kernel(
    const float* __restrict__ z_re,
    const float* __restrict__ z_im,
    float* __restrict__ out, int B) {
  __shared__ __align__(16) float2 sm_z [ROWS_PER_BLOCK][5];
  __shared__ __align__(16) float2 sm_zz[ROWS_PER_BLOCK][NTRI];

  const int t    = threadIdx.x;
  const int w    = t >> 5;        // wave id; handles local rows 2w, 2w+1
  const int lane = t & 31;
  const int r0   = blockIdx.x * ROWS_PER_BLOCK;

  // ---- Phase 1: async global -> LDS staging of this block's z values ------
  // 80 re + 80 im floats; one b32 async copy per participating thread,
  // interleaving re into .x and im into .y.
  const int rem    = B - r0;
  const int nrows  = rem >= ROWS_PER_BLOCK ? ROWS_PER_BLOCK : rem;
  const int nvalid = nrows * 5;                    // <= 80
  if (t < nvalid) {
    const float* gp = z_re + (size_t)r0 * 5 + t;
    lds_float* dp = (lds_float*)&sm_z[t / 5][t % 5].x;
    asm volatile("global_load_async_to_lds_b32 %0, %1, off"
                 :: "v"(dp), "v"(gp) : "memory");
  } else if (t >= 80 && (t - 80) < nvalid) {
    const int u = t - 80;
    const float* gp = z_im + (size_t)r0 * 5 + u;
    lds_float* dp = (lds_float*)&sm_z[u / 5][u % 5].y;
    asm volatile("global_load_async_to_lds_b32 %0, %1, off"
                 :: "v"(dp), "v"(gp) : "memory");
  }
  asm volatile("s_wait_asynccnt 0x0" ::: "memory");
  __syncthreads();

  const int row0 = r0 + 2 * w;

  // ---- Phase 2: each wave builds 2 rows x 35 monomials into LDS -----------
  if (row0 < B) {
#pragma unroll
    for (int chunk = 0; chunk < 3; ++chunk) {
      const int m = lane + chunk * 32;             // 0..95, active m < 70
      if (m < 2 * NTRI) {
        const int  sub = (m >= NTRI) ? 1 : 0;      // which of the two rows
        const int  mi  = m - NTRI * sub;
        const int  lr  = 2 * w + sub;
        const unsigned c = TRI.v[mi];
        const float2 a  = sm_z[lr][c & 15];
        const float2 b  = sm_z[lr][(c >> 4) & 15];
        const float2 cc = sm_z[lr][c >> 8];
        // (a*b) left-associated, matching the reference's evaluation order.
        const float tr = a.x * b.x - a.y * b.y;
        const float ti = a.x * b.y + a.y * b.x;
        float2 zz;
        zz.x = tr * cc.x - ti * cc.y;
        zz.y = tr * cc.y + ti * cc.x;
        sm_zz[lr][mi] = zz;
      }
    }
  }
  __syncthreads();

  // ---- Phase 3: dense flattened column sweep, 2 rows per wave -------------
  if (row0 < B) {
    float* outRow = out + (size_t)row0 * NCOL;
    const char* zzb = (const char*)&sm_zz[2 * w][0];
    if (row0 + 1 < B) sweep<true >(zzb, outRow, lane);
    else              sweep<false>(zzb, outRow, lane);
  }
}

extern "C" void kernel_launch(void* const* d_in, const int* in_sizes, int n_in,
                              void* d_out, int out_size, void* d_ws, size_t ws_size,
                              hipStream_t stream) {
  (void)n_in; (void)out_size; (void)d_ws; (void)ws_size;
  const float* z_re = (const float*)d_in[0];
  const float* z_im = (const float*)d_in[1];
  float* out = (float*)d_out;
  const int B = in_sizes[0] / 5;  // (B, 5) f32
  const int blocks = (B + ROWS_PER_BLOCK - 1) / ROWS_PER_BLOCK;
  bihom_k3_kernel<<<blocks, 256, 0, stream>>>(z_re, z_im, out, B);
}